// CRF_24404004176103
// MI455X (gfx1250) — compile-verified
//
#include <hip/hip_runtime.h>
#include <hip/hip_bf16.h>

// CRF NLL, B=128, S=2048, T=64.
// Forward recurrence reformulated as fp32 WMMA GEMM:
//   alpha'_j = m + c_j + log( sum_i exp(alpha_i - m) * exp(trans_ij - c_j) ) + em_tj
// 8 blocks x 128 threads; each block owns 16 batch rows; wave w owns 16 output cols.
// exp(trans - cmax) B-fragments live in registers for the whole 2048-step scan.

typedef __attribute__((ext_vector_type(2))) float v2f;
typedef __attribute__((ext_vector_type(8))) float v8f;

#define CRF_B 128
#define CRF_S 2048
#define CRF_T 64

__global__ void __launch_bounds__(256)
crf_score_kernel(const float* __restrict__ em, const int* __restrict__ tags,
                 const float* __restrict__ trans, const float* __restrict__ start,
                 const float* __restrict__ endt, float* __restrict__ score) {
    const int b = blockIdx.x;
    const int tid = threadIdx.x;
    __shared__ float red[256];
    const int* tb = tags + (size_t)b * CRF_S;
    float s = 0.0f;
    for (int t = tid + 1; t < CRF_S; t += 256) {
        const int prev = tb[t - 1];
        const int cur  = tb[t];
        s += trans[prev * CRF_T + cur];
        s += em[((size_t)b * CRF_S + t) * CRF_T + cur];
    }
    red[tid] = s;
    __syncthreads();
    for (int off = 128; off > 0; off >>= 1) {
        if (tid < off) red[tid] += red[tid + off];
        __syncthreads();
    }
    if (tid == 0) score[b] = red[0] + start[tb[0]] + endt[tb[CRF_S - 1]];
}

__global__ void __launch_bounds__(128)
crf_forward_kernel(const float* __restrict__ em, const float* __restrict__ trans,
                   const float* __restrict__ start, const float* __restrict__ endt,
                   float* __restrict__ logz) {
    __shared__ float alpha[16][CRF_T + 1];  // current alpha, 16 batch rows x 64 states
    __shared__ float pbuf [16][CRF_T + 1];  // exp(alpha - rowmax)
    __shared__ float mrow[16];              // per-row max
    __shared__ float cmax[CRF_T];           // per-column max of transitions

    const int tid   = threadIdx.x;
    const int wave  = tid >> 5;             // 0..3: which 16-col output tile
    const int lane  = tid & 31;
    const int khalf = lane >> 4;            // 0 or 1 (K-halves in A/B fragments)
    const int ncol  = (wave << 4) + (lane & 15);  // this lane's output column (0..63)
    const int b0    = blockIdx.x * 16;      // first batch row of this block

    // ---- column maxes of transitions ----
    if (tid < CRF_T) {
        float c = trans[tid];
        for (int i = 1; i < CRF_T; ++i) c = fmaxf(c, trans[i * CRF_T + tid]);
        cmax[tid] = c;
    }
    // ---- alpha0 = start + emissions[:, 0, :] ----
    for (int idx = tid; idx < 16 * CRF_T; idx += 128) {
        const int r = idx >> 6, c = idx & 63;
        alpha[r][c] = start[c] + em[((size_t)(b0 + r) * CRF_S) * CRF_T + c];
    }
    __syncthreads();

    // ---- loop-invariant B fragments: exp(trans[k][ncol] - cmax[ncol]) ----
    // Layout mirror of C/D & A striping: k-slice kk, vgpr u, lane L:
    //   K = 4*kk + 2*(L>=16) + u, N = L%16.
    const float cj = cmax[ncol];
    v2f bfrag[16];
#pragma unroll
    for (int kk = 0; kk < 16; ++kk) {
        const int kbase = 4 * kk + 2 * khalf;
        bfrag[kk].x = __expf(trans[(kbase + 0) * CRF_T + ncol] - cj);
        bfrag[kk].y = __expf(trans[(kbase + 1) * CRF_T + ncol] - cj);
    }

    // ---- prefetch emissions for t=1 into registers ----
    // C/D layout: vgpr v, lane L -> row M = v + 8*(L>=16), col N = L%16.
    float emc[8];
#pragma unroll
    for (int v = 0; v < 8; ++v) {
        const int M = v + 8 * khalf;
        emc[v] = em[((size_t)(b0 + M) * CRF_S + 1) * CRF_T + ncol];
    }

    // ---- sequential scan over timesteps ----
    for (int t = 1; t < CRF_S; ++t) {
        // Phase 1: row maxes + p = exp(alpha - m).  8 threads per row.
        const int r = tid >> 3, g = tid & 7;
        float av[8];
        av[0] = alpha[r][g];
        float locmax = av[0];
#pragma unroll
        for (int q = 1; q < 8; ++q) {
            av[q] = alpha[r][g + 8 * q];
            locmax = fmaxf(locmax, av[q]);
        }
#pragma unroll
        for (int off = 1; off < 8; off <<= 1)
            locmax = fmaxf(locmax, __shfl_xor(locmax, off, 8));
#pragma unroll
        for (int q = 0; q < 8; ++q) pbuf[r][g + 8 * q] = __expf(av[q] - locmax);
        if (g == 0) mrow[r] = locmax;
        __syncthreads();

        // Prefetch emissions for t+1 (hidden under WMMA).
        const int tn = (t + 1 < CRF_S) ? (t + 1) : t;
        float emn[8];
#pragma unroll
        for (int v = 0; v < 8; ++v) {
            const int M = v + 8 * khalf;
            emn[v] = em[((size_t)(b0 + M) * CRF_S + tn) * CRF_T + ncol];
        }

        // Phase 2: 16x16 tile = P(16x64) * expT(64x16) via 16 chained f32 WMMAs,
        // two accumulators to shorten the C dependency chain.
        v8f acc0 = {};
        v8f acc1 = {};
        const int arow = lane & 15;  // A: lanes 0-15 -> M=lane, K=0,1; lanes 16-31 -> K=2,3
#pragma unroll
        for (int kk = 0; kk < 16; ++kk) {
            const int kbase = 4 * kk + 2 * khalf;
            v2f a;
            a.x = pbuf[arow][kbase + 0];
            a.y = pbuf[arow][kbase + 1];
            if (kk & 1)
                acc1 = __builtin_amdgcn_wmma_f32_16x16x4_f32(
                    false, a, false, bfrag[kk], (short)0, acc1, false, false);
            else
                acc0 = __builtin_amdgcn_wmma_f32_16x16x4_f32(
                    false, a, false, bfrag[kk], (short)0, acc0, false, false);
        }
        const v8f acc = acc0 + acc1;

        // Epilogue: alpha'[M][ncol] = m[M] + c_j + log(s) + em[t]
#pragma unroll
        for (int v = 0; v < 8; ++v) {
            const int M = v + 8 * khalf;
            alpha[M][ncol] = mrow[M] + cj + __logf(acc[v]) + emc[v];
        }
#pragma unroll
        for (int v = 0; v < 8; ++v) emc[v] = emn[v];
        __syncthreads();
    }

    // ---- log_z = logsumexp_j(alpha + end) per batch row ----
    {
        const int r = tid >> 3, g = tid & 7;
        float vals[8];
        vals[0] = alpha[r][g] + endt[g];
        float mx = vals[0];
#pragma unroll
        for (int q = 1; q < 8; ++q) {
            vals[q] = alpha[r][g + 8 * q] + endt[g + 8 * q];
            mx = fmaxf(mx, vals[q]);
        }
#pragma unroll
        for (int off = 1; off < 8; off <<= 1)
            mx = fmaxf(mx, __shfl_xor(mx, off, 8));
        float sum = 0.0f;
#pragma unroll
        for (int q = 0; q < 8; ++q) sum += __expf(vals[q] - mx);
#pragma unroll
        for (int off = 1; off < 8; off <<= 1)
            sum += __shfl_xor(sum, off, 8);
        if (g == 0) logz[b0 + r] = mx + __logf(sum);
    }
}

__global__ void __launch_bounds__(128)
crf_final_kernel(const float* __restrict__ score, const float* __restrict__ logz,
                 float* __restrict__ out) {
    __shared__ float red[128];
    const int tid = threadIdx.x;
    red[tid] = logz[tid] - score[tid];
    __syncthreads();
    for (int off = 64; off > 0; off >>= 1) {
        if (tid < off) red[tid] += red[tid + off];
        __syncthreads();
    }
    if (tid == 0) out[0] = red[0] * (1.0f / (float)CRF_B);
}

extern "C" void kernel_launch(void* const* d_in, const int* in_sizes, int n_in,
                              void* d_out, int out_size, void* d_ws, size_t ws_size,
                              hipStream_t stream) {
    (void)in_sizes; (void)n_in; (void)out_size; (void)ws_size;
    const float* em    = (const float*)d_in[0];   // (128, 2048, 64) f32
    const int*   tags  = (const int*)  d_in[1];   // (128, 2048) i32
    const float* trans = (const float*)d_in[2];   // (64, 64) f32
    const float* start = (const float*)d_in[3];   // (64,) f32
    const float* endt  = (const float*)d_in[4];   // (64,) f32
    float* out = (float*)d_out;

    float* score = (float*)d_ws;        // 128 floats
    float* logz  = score + CRF_B;       // 128 floats

    crf_score_kernel<<<CRF_B, 256, 0, stream>>>(em, tags, trans, start, endt, score);
    crf_forward_kernel<<<CRF_B / 16, 128, 0, stream>>>(em, trans, start, endt, logz);
    crf_final_kernel<<<1, CRF_B, 0, stream>>>(score, logz, out);
}